// Sampler_14886356648673
// MI455X (gfx1250) — compile-verified
//
#include <hip/hip_runtime.h>
#include <math.h>

typedef float v4f __attribute__((ext_vector_type(4)));

#define WG    256      // 8 wave32 waves per workgroup
#define SPLIT 4        // workgroups per row -> 1024 WGs total for latency hiding
#define NWAVES (WG / 32)

// ---- argmax combine rules -------------------------------------------------
// Within one thread the scan index is strictly increasing, so a strict '>'
// keeps the earliest index on ties (matches jnp.argmax first-occurrence).
__device__ __forceinline__ void take_scan(float v, int i, float& bv, int& bi) {
  if (v > bv) { bv = v; bi = i; }
}
// Across threads/waves/parts: full tie rule (max value, then min index).
// Commutative + associative, safe for xor-butterfly reduction.
__device__ __forceinline__ void take_merge(float v, int i, float& bv, int& bi) {
  if (v > bv || (v == bv && i < bi)) { bv = v; bi = i; }
}

// t = -log1p(-n) = -log(1-n), n in [0,1).
// Small n: 5-term series (rel err < 1e-5 for n < 0.125); n==0 -> t==0 so the
// outer log gives -inf and key = +inf, matching probs/0 = inf in the reference.
// Large n: hardware v_log_f32 on (1-n) <= 0.875, well away from the arg~1 region.
__device__ __forceinline__ float neg_log1p_neg(float n) {
  float p = fmaf(n, 0.20f, 0.25f);
  p = fmaf(n, p, 0.33333333f);
  p = fmaf(n, p, 0.5f);
  p = fmaf(n, p, 1.0f);
  float t_small = n * p;
  float t_big = -0.69314718f * __log2f(1.0f - n);   // v_log_f32
  return (n < 0.125f) ? t_small : t_big;
}

__global__ __launch_bounds__(WG) void sampler_partial(
    const float* __restrict__ logits,
    const float* __restrict__ temps,
    const float* __restrict__ noise,
    float* __restrict__ wsVal,
    int*   __restrict__ wsIdx,
    int V)
{
  const int row  = blockIdx.x / SPLIT;
  const int part = blockIdx.x % SPLIT;
  const int tid  = threadIdx.x;

  const float temp  = temps[row];
  const size_t rbase = (size_t)row * (size_t)V;
  const float* rowL = logits + rbase;
  const float* rowN = noise  + rbase;

  const int nvec  = V >> 2;
  const int chunk = (nvec + SPLIT - 1) / SPLIT;
  const int vbeg  = part * chunk;
  const int vend  = min(nvec, vbeg + chunk);

  float bv = -INFINITY;
  int   bi = 0x7fffffff;

  if (temp == 0.0f) {
    // Greedy row: stream only logits (saves the noise read entirely).
    const v4f* pL = (const v4f*)rowL;
    #pragma unroll 4
    for (int i = vbeg + tid; i < vend; i += WG) {
      v4f L = __builtin_nontemporal_load(pL + i);       // global_load_b128 NT
      __builtin_prefetch(pL + i + 8 * WG, 0, 0);        // global_prefetch_b8
      const int b = i << 2;
      take_scan(L.x, b,     bv, bi);
      take_scan(L.y, b + 1, bv, bi);
      take_scan(L.z, b + 2, bv, bi);
      take_scan(L.w, b + 3, bv, bi);
    }
  } else {
    // Sampled row: Gumbel-max key = logit/T - log(-log1p(-noise)).
    const float invT = 1.0f / temp;
    const v4f* pL = (const v4f*)rowL;
    const v4f* pN = (const v4f*)rowN;
    #pragma unroll 4
    for (int i = vbeg + tid; i < vend; i += WG) {
      v4f L = __builtin_nontemporal_load(pL + i);
      v4f N = __builtin_nontemporal_load(pN + i);
      __builtin_prefetch(pL + i + 8 * WG, 0, 0);
      __builtin_prefetch(pN + i + 8 * WG, 0, 0);
      const int b = i << 2;
      {
        float g = 0.69314718f * __log2f(neg_log1p_neg(N.x));
        take_scan(fmaf(L.x, invT, -g), b,     bv, bi);
      }
      {
        float g = 0.69314718f * __log2f(neg_log1p_neg(N.y));
        take_scan(fmaf(L.y, invT, -g), b + 1, bv, bi);
      }
      {
        float g = 0.69314718f * __log2f(neg_log1p_neg(N.z));
        take_scan(fmaf(L.z, invT, -g), b + 2, bv, bi);
      }
      {
        float g = 0.69314718f * __log2f(neg_log1p_neg(N.w));
        take_scan(fmaf(L.w, invT, -g), b + 3, bv, bi);
      }
    }
  }

  // Scalar tail if V % 4 != 0 (not hit for V = 128256, kept for generality).
  if (part == SPLIT - 1) {
    const float invT = (temp == 0.0f) ? 0.0f : 1.0f / temp;
    for (int v = (nvec << 2) + tid; v < V; v += WG) {
      float l = rowL[v];
      if (temp == 0.0f) {
        take_scan(l, v, bv, bi);
      } else {
        float g = 0.69314718f * __log2f(neg_log1p_neg(rowN[v]));
        take_scan(fmaf(l, invT, -g), v, bv, bi);
      }
    }
  }

  // Wave32 xor-butterfly reduction (ds_swizzle / permute path on gfx1250).
  #pragma unroll
  for (int off = 16; off > 0; off >>= 1) {
    float ov = __shfl_xor(bv, off, 32);
    int   oi = __shfl_xor(bi, off, 32);
    take_merge(ov, oi, bv, bi);
  }

  __shared__ float sV[NWAVES];
  __shared__ int   sI[NWAVES];
  const int wave = tid >> 5;
  const int lane = tid & 31;
  if (lane == 0) { sV[wave] = bv; sI[wave] = bi; }
  __syncthreads();
  if (tid == 0) {
    float fv = sV[0]; int fi = sI[0];
    #pragma unroll
    for (int w = 1; w < NWAVES; ++w) take_merge(sV[w], sI[w], fv, fi);
    wsVal[blockIdx.x] = fv;   // blockIdx.x == row*SPLIT + part
    wsIdx[blockIdx.x] = fi;
  }
}

__global__ void sampler_combine(const float* __restrict__ wsVal,
                                const int*   __restrict__ wsIdx,
                                float* __restrict__ out, int B)
{
  int r = blockIdx.x * blockDim.x + threadIdx.x;
  if (r >= B) return;
  float bv = wsVal[r * SPLIT];
  int   bi = wsIdx[r * SPLIT];
  #pragma unroll
  for (int p = 1; p < SPLIT; ++p)
    take_merge(wsVal[r * SPLIT + p], wsIdx[r * SPLIT + p], bv, bi);
  out[r] = (float)bi;   // token index, exactly representable in fp32
}

extern "C" void kernel_launch(void* const* d_in, const int* in_sizes, int n_in,
                              void* d_out, int out_size, void* d_ws, size_t ws_size,
                              hipStream_t stream) {
  const float* logits = (const float*)d_in[0];
  const float* temps  = (const float*)d_in[1];
  const float* noise  = (const float*)d_in[2];
  const int B = in_sizes[1];
  const int V = in_sizes[0] / B;

  float* wsVal = (float*)d_ws;
  int*   wsIdx = (int*)((char*)d_ws + (size_t)B * SPLIT * sizeof(float));

  sampler_partial<<<B * SPLIT, WG, 0, stream>>>(logits, temps, noise,
                                                wsVal, wsIdx, V);
  const int cb = 256;
  sampler_combine<<<(B + cb - 1) / cb, cb, 0, stream>>>(wsVal, wsIdx,
                                                        (float*)d_out, B);
}